// OuterEquivariantMap_16080357556715
// MI455X (gfx1250) — compile-verified
//
#include <hip/hip_runtime.h>

typedef float v2f __attribute__((ext_vector_type(2)));
typedef float v8f __attribute__((ext_vector_type(8)));
typedef unsigned int v4u __attribute__((ext_vector_type(4)));
typedef int v4i __attribute__((ext_vector_type(4)));
typedef int v8i __attribute__((ext_vector_type(8)));

#define KVOX  200000
#define NPTS  1000000
#define NOFF  27
#define OUT_C 64
#define KC    81     // 27*3
#define KCP   84     // padded to multiple of 4
#define KL    67     // 3 + 64
#define KLP   68     // padded to multiple of 4

// ---------------- TDM staging: contiguous 1-D f32 block, global -> LDS ----------------
// D# descriptor per CDNA5 ISA ch.8 (tensor as 1-D row: tile_dim0 = tensor_dim0 = nelem).
// This toolchain exposes the 6-arg builtin: (v4u, v8i, v4i, v4i, v8i, i32 cpol).

#if __has_builtin(__builtin_amdgcn_tensor_load_to_lds)
#define HAS_TDM 1
__device__ __forceinline__ void tdm_load_1d_f32(unsigned lds_off, const float* gptr,
                                                unsigned nelem) {
    unsigned long long ga = (unsigned long long)(uintptr_t)gptr;
    v4u g0;
    g0.x = 1u;                                               // count=1, user descriptor
    g0.y = lds_off;                                          // lds_addr (bytes)
    g0.z = (unsigned)(ga & 0xffffffffu);                     // global_addr[31:0]
    g0.w = (unsigned)((ga >> 32) & 0x01ffffffu) | (2u << 30); // global_addr[56:32] | type=2
    v8i g1;
    g1[0] = (int)(2u << 16);                                 // data_size = 4B
    g1[1] = (int)((nelem & 0xffffu) << 16);                  // tensor_dim0[15:0]
    g1[2] = (int)((nelem >> 16) & 0xffffu) | (1 << 16);      // tensor_dim0[31:16] | tensor_dim1=1
    g1[3] = (int)((nelem & 0xffffu) << 16);                  // tile_dim0 = nelem
    g1[4] = 0;                                               // tile_dim1/2 unused
    g1[5] = (int)nelem;                                      // tensor_dim0_stride[31:0]
    g1[6] = 0;
    g1[7] = 0;
    v4i g2; g2[0] = 1; g2[1] = 1; g2[2] = 0; g2[3] = 0;      // tensor_dim2=1, tensor_dim3=1
    v4i g3; g3[0] = 0; g3[1] = 0; g3[2] = 0; g3[3] = 0;
    v8i g4; g4[0] = 0; g4[1] = 0; g4[2] = 0; g4[3] = 0;
            g4[4] = 0; g4[5] = 0; g4[6] = 0; g4[7] = 0;
    __builtin_amdgcn_tensor_load_to_lds(g0, g1, g2, g3, g4, 0);
}
#endif

// ---------------- helper kernels ----------------

__global__ void zero_kernel(float* __restrict__ p, int n) {
    int i = blockIdx.x * blockDim.x + threadIdx.x;
    if (i < n) p[i] = 0.0f;
}

__global__ void scatter_kernel(const float* __restrict__ x,
                               const int* __restrict__ cluster,
                               float* __restrict__ sums,
                               float* __restrict__ cnt, int n) {
    int i = blockIdx.x * blockDim.x + threadIdx.x;
    if (i >= n) return;
    int c = cluster[i];
    atomicAdd(&sums[c * 3 + 0], x[i * 3 + 0]);
    atomicAdd(&sums[c * 3 + 1], x[i * 3 + 1]);
    atomicAdd(&sums[c * 3 + 2], x[i * 3 + 2]);
    atomicAdd(&cnt[c], 1.0f);
}

__global__ void pooled_kernel(const float* __restrict__ sums,
                              const float* __restrict__ cnt,
                              float* __restrict__ pooled, int k) {
    int i = blockIdx.x * blockDim.x + threadIdx.x;
    if (i >= k) return;
    float c = fmaxf(cnt[i], 1.0f);
    float inv = 1.0f / c;
    pooled[i * 4 + 0] = sums[i * 3 + 0] * inv;
    pooled[i * 4 + 1] = sums[i * 3 + 1] * inv;
    pooled[i * 4 + 2] = sums[i * 3 + 2] * inv;
    pooled[i * 4 + 3] = 0.0f;
}

// ---------------- conv GEMM: (K x 81) @ (81 x 64) via f32 WMMA ----------------
// 128 threads = 4 waves; block covers 64 voxels; each wave one 16x64 tile.
// W_conv staged into LDS by the Tensor Data Mover while lanes gather g.

__global__ __launch_bounds__(128) void conv_wmma(const int* __restrict__ nb,
                                                 const float* __restrict__ Wc,
                                                 const float* __restrict__ pooled,
                                                 float* __restrict__ outf) {
    __shared__ float sW[KCP * OUT_C];   // 84 x 64, rows 81..83 zero
    __shared__ float sG[64 * KCP];      // 64 voxels x 84, cols 81..83 zero

    const int tid  = threadIdx.x;
    const int base = blockIdx.x * 64;

#ifdef HAS_TDM
    // wave 0 issues the DMA for W_conv (81x64 contiguous f32)
    if (tid < 32) {
        tdm_load_1d_f32((unsigned)(uintptr_t)&sW[0], Wc, KC * OUT_C);
    }
    // zero-pad K rows 81..83 while the DMA runs
    for (int idx = tid; idx < (KCP - KC) * OUT_C; idx += 128) {
        sW[KC * OUT_C + idx] = 0.0f;
    }
#else
    for (int idx = tid; idx < KCP * OUT_C; idx += 128) {
        int r = idx / OUT_C;
        sW[idx] = (r < KC) ? Wc[idx] : 0.0f;
    }
#endif

    // Gather g into LDS: g[v][o*3+i] = nb>=0 ? pooled[nb][i] : 0
    for (int idx = tid; idx < 64 * KCP; idx += 128) {
        int v = idx / KCP;
        int j = idx - v * KCP;
        float val = 0.0f;
        if (j < KC) {
            int o = j / 3;
            int i = j - o * 3;
            int n = nb[(base + v) * NOFF + o];
            if (n >= 0) val = pooled[n * 4 + i];
        }
        sG[idx] = val;
    }

#ifdef HAS_TDM
    if (tid < 32) __builtin_amdgcn_s_wait_tensorcnt(0);
#endif
    __syncthreads();

    const int lane = tid & 31;
    const int wv   = tid >> 5;            // wave id 0..3
    const int row  = lane & 15;           // A: M row / B,D: N column
    const int koff = (lane >> 4) << 1;    // 0 for lanes 0-15, 2 for 16-31
    const float* gRow = &sG[(wv * 16 + row) * KCP];

    v8f acc[4];
    acc[0] = v8f{}; acc[1] = v8f{}; acc[2] = v8f{}; acc[3] = v8f{};

    for (int kk = 0; kk < KCP; kk += 4) {
        v2f a;
        a.x = gRow[kk + koff];
        a.y = gRow[kk + koff + 1];
        const float* w0 = &sW[(kk + koff) * OUT_C + row];
        const float* w1 = w0 + OUT_C;
#pragma unroll
        for (int ct = 0; ct < 4; ++ct) {
            v2f b;
            b.x = w0[ct * 16];
            b.y = w1[ct * 16];
            acc[ct] = __builtin_amdgcn_wmma_f32_16x16x4_f32(
                false, a, false, b, (short)0, acc[ct], false, false);
        }
    }

    // D layout: lanes 0-15 -> M=r, N=lane; lanes 16-31 -> M=r+8, N=lane-16
    const int up   = (lane >> 4) << 3;    // 0 or 8
    const int mrow = base + wv * 16 + up;
#pragma unroll
    for (int r = 0; r < 8; ++r) {
        float* dst = &outf[(size_t)(mrow + r) * OUT_C + row];
        dst[0]  = acc[0][r];   // out_f stays RT: re-read by linear phase, L2-resident
        dst[16] = acc[1][r];
        dst[32] = acc[2][r];
        dst[48] = acc[3][r];
    }
}

// ------------- linear GEMM: (N x 67) @ (67 x 64) + bias via f32 WMMA -------------

__global__ __launch_bounds__(128) void linear_wmma(const float* __restrict__ x,
                                                   const int* __restrict__ cluster,
                                                   const float* __restrict__ Wl,
                                                   const float* __restrict__ bl,
                                                   const float* __restrict__ outf,
                                                   float* __restrict__ out) {
    __shared__ float sW[KLP * OUT_C];   // 68 x 64, row 67 zero
    __shared__ float sH[64 * KLP];      // 64 points x 68, col 67 zero
    __shared__ int   sC[64];

    const int tid  = threadIdx.x;
    const int base = blockIdx.x * 64;

#ifdef HAS_TDM
    if (tid < 32) {
        tdm_load_1d_f32((unsigned)(uintptr_t)&sW[0], Wl, KL * OUT_C);
    }
    for (int idx = tid; idx < (KLP - KL) * OUT_C; idx += 128) {
        sW[KL * OUT_C + idx] = 0.0f;
    }
#else
    for (int idx = tid; idx < KLP * OUT_C; idx += 128) {
        int r = idx / OUT_C;
        sW[idx] = (r < KL) ? Wl[idx] : 0.0f;
    }
#endif

    // stage cluster ids once
    if (tid < 64) sC[tid] = cluster[base + tid];
    __syncthreads();

    // h[v] = [ x[p,0..2] | outf[cluster[p], 0..63] | 0 ]
    for (int idx = tid; idx < 64 * KLP; idx += 128) {
        int v = idx / KLP;
        int j = idx - v * KLP;
        float val = 0.0f;
        if (j < 3) {
            val = x[(size_t)(base + v) * 3 + j];
        } else if (j < KL) {
            val = outf[(size_t)sC[v] * OUT_C + (j - 3)];
        }
        sH[idx] = val;
    }

#ifdef HAS_TDM
    if (tid < 32) __builtin_amdgcn_s_wait_tensorcnt(0);
#endif
    __syncthreads();

    const int lane = tid & 31;
    const int wv   = tid >> 5;
    const int row  = lane & 15;
    const int koff = (lane >> 4) << 1;
    const float* hRow = &sH[(wv * 16 + row) * KLP];

    v8f acc[4];
    acc[0] = v8f{}; acc[1] = v8f{}; acc[2] = v8f{}; acc[3] = v8f{};

    for (int kk = 0; kk < KLP; kk += 4) {
        v2f a;
        a.x = hRow[kk + koff];
        a.y = hRow[kk + koff + 1];
        const float* w0 = &sW[(kk + koff) * OUT_C + row];
        const float* w1 = w0 + OUT_C;
#pragma unroll
        for (int ct = 0; ct < 4; ++ct) {
            v2f b;
            b.x = w0[ct * 16];
            b.y = w1[ct * 16];
            acc[ct] = __builtin_amdgcn_wmma_f32_16x16x4_f32(
                false, a, false, b, (short)0, acc[ct], false, false);
        }
    }

    const float b0 = bl[row];
    const float b1 = bl[row + 16];
    const float b2 = bl[row + 32];
    const float b3 = bl[row + 48];

    const int up   = (lane >> 4) << 3;
    const int mrow = base + wv * 16 + up;
#pragma unroll
    for (int r = 0; r < 8; ++r) {
        float* dst = &out[(size_t)(mrow + r) * OUT_C + row];
        // final output (256 MB) is write-once: non-temporal to preserve L2 for outf/W
        __builtin_nontemporal_store(acc[0][r] + b0, dst + 0);
        __builtin_nontemporal_store(acc[1][r] + b1, dst + 16);
        __builtin_nontemporal_store(acc[2][r] + b2, dst + 32);
        __builtin_nontemporal_store(acc[3][r] + b3, dst + 48);
    }
}

// ---------------- launcher ----------------

extern "C" void kernel_launch(void* const* d_in, const int* in_sizes, int n_in,
                              void* d_out, int out_size, void* d_ws, size_t ws_size,
                              hipStream_t stream) {
    const float* x       = (const float*)d_in[0];
    const int*   cluster = (const int*)d_in[1];
    const int*   nb      = (const int*)d_in[2];
    const float* Wc      = (const float*)d_in[3];
    const float* Wl      = (const float*)d_in[4];
    const float* bl      = (const float*)d_in[5];
    float* out = (float*)d_out;

    // workspace layout (floats):
    //   [0, 3K)      segment sums
    //   [3K, 4K)     segment counts
    //   [4K, 8K)     pooled means (stride 4, padded)
    //   [8K, 8K+64K) out_f (K x 64)
    float* ws     = (float*)d_ws;
    float* sums   = ws;
    float* cnt    = ws + (size_t)KVOX * 3;
    float* pooled = ws + (size_t)KVOX * 4;
    float* outf   = ws + (size_t)KVOX * 8;

    zero_kernel<<<(KVOX * 4 + 255) / 256, 256, 0, stream>>>(sums, KVOX * 4);
    scatter_kernel<<<(NPTS + 255) / 256, 256, 0, stream>>>(x, cluster, sums, cnt, NPTS);
    pooled_kernel<<<(KVOX + 255) / 256, 256, 0, stream>>>(sums, cnt, pooled, KVOX);

    // 200000 / 64 = 3125 exactly; 1000000 / 64 = 15625 exactly (no tails)
    conv_wmma<<<KVOX / 64, 128, 0, stream>>>(nb, Wc, pooled, outf);
    linear_wmma<<<NPTS / 64, 128, 0, stream>>>(x, cluster, Wl, bl, outf, out);
}